// SubGL_78975858639097
// MI455X (gfx1250) — compile-verified
//
#include <hip/hip_runtime.h>

typedef __attribute__((ext_vector_type(2))) float v2f;
typedef __attribute__((ext_vector_type(8))) float v8f;

#define B_  64
#define L_  200
#define N_  32
#define D_  128
#define WPB 8   // waves per block

__global__ __launch_bounds__(256) void subgl_agg_kernel(
    const int*   __restrict__ sequence,      // [B,L]
    const int*   __restrict__ seq_neighbor,  // [B,L,N]
    const int*   __restrict__ rel_neigh,     // [B,L,N]
    const float* __restrict__ user_emb,      // [USER_NUM, D]
    const float* __restrict__ rel_emb,       // [3,3]
    const float* __restrict__ weight_b,      // [3,1]
    float*       __restrict__ out)           // [B,L,D]
{
    __shared__ float sm[WPB * D_];
    const int lane = threadIdx.x & 31;
    const int wid  = threadIdx.x >> 5;
    const int bl   = blockIdx.x * WPB + wid;        // 0 .. B*L-1 (grid is exact)
    if (bl >= B_ * L_) return;                      // wave-uniform; never taken
    const int h = lane >> 4;                        // half-wave: selects K-pair of A
    const int m = lane & 15;                        // M index of A (row = dim)
    float* wsm = sm + wid * D_;

    // beta = softmax(weight_b) over the 3 relations (redundant per lane; 3 floats)
    const float q0 = weight_b[0], q1 = weight_b[1], q2 = weight_b[2];
    const float mx = fmaxf(q0, fmaxf(q1, q2));
    const float e0 = expf(q0 - mx), e1 = expf(q1 - mx), e2 = expf(q2 - mx);
    const float rs = 1.0f / (e0 + e1 + e2);
    const float b0 = e0 * rs, b1 = e1 * rs, b2 = e2 * rs;

    // lane n owns neighbor n: its index and mixing weight w[n]
    const int   nb   = seq_neighbor[(long)bl * N_ + lane];
    const int   rl   = rel_neigh  [(long)bl * N_ + lane];
    const float* rr  = rel_emb + rl * 3;
    const float w    = rr[0] * b0 + rr[1] * b1 + rr[2] * b2;

    // Pre-broadcast the per-group (4 neighbors) weights / row offsets this
    // lane-half needs for the A-matrix:  VGPR0 <- K=2h, VGPR1 <- K=2h+1.
    long  rowA[8], rowB[8];
    float wA[8],  wB[8];
#pragma unroll
    for (int g = 0; g < 8; ++g) {
        const int k0 = g * 4 + 2 * h;
        wA[g]   = __shfl(w,  k0,     32);
        wB[g]   = __shfl(w,  k0 + 1, 32);
        rowA[g] = (long)__shfl(nb, k0,     32) * D_;
        rowB[g] = (long)__shfl(nb, k0 + 1, 32) * D_;
    }

    const v2f bones = {1.0f, 1.0f};   // B = all-ones: layout-independent column replication

    // agg[16t + m] = sum over 32 neighbors, computed as 8 chained f32 WMMAs per tile.
#pragma unroll
    for (int t = 0; t < 8; ++t) {
        v8f c = {0.f, 0.f, 0.f, 0.f, 0.f, 0.f, 0.f, 0.f};
        const int dim = t * 16 + m;
#pragma unroll
        for (int g = 0; g < 8; ++g) {
            v2f a;
            a.x = wA[g] * user_emb[rowA[g] + dim];   // A[m][2h]   pre-scaled by w
            a.y = wB[g] * user_emb[rowB[g] + dim];   // A[m][2h+1] pre-scaled by w
            // D[m][n] = sum_k A[m][k] * 1 + C[m][n]
            c = __builtin_amdgcn_wmma_f32_16x16x4_f32(
                    false, a, false, bones, (short)0, c, false, false);
        }
        // C/D layout: VGPR r holds M=r (lanes 0-15) / M=8+r (lanes 16-31); all N equal.
        if (m == 0) {
            float* p = wsm + t * 16 + 8 * h;
            *(float4*)(p)     = make_float4(c[0], c[1], c[2], c[3]);
            *(float4*)(p + 4) = make_float4(c[4], c[5], c[6], c[7]);
        }
    }

    // Same-wave LDS is in-order; make the cross-lane readback explicit anyway.
    asm volatile("s_wait_dscnt 0" ::: "memory");

    const float4 agg = *(const float4*)(wsm + 4 * lane);
    const long   s   = (long)sequence[bl];
    const float4 se  = *(const float4*)(user_emb + s * D_ + 4 * lane);
    float cx = agg.x + se.x, cy = agg.y + se.y, cz = agg.z + se.z, cw = agg.w + se.w;

    float ss = cx * cx + cy * cy + cz * cz + cw * cw;
#pragma unroll
    for (int off = 16; off; off >>= 1) ss += __shfl_xor(ss, off, 32);
    const float scale = 1.0f / fmaxf(sqrtf(ss), 1e-12f);

    *(float4*)(out + (long)bl * D_ + 4 * lane) =
        make_float4(cx * scale, cy * scale, cz * scale, cw * scale);
}

extern "C" void kernel_launch(void* const* d_in, const int* in_sizes, int n_in,
                              void* d_out, int out_size, void* d_ws, size_t ws_size,
                              hipStream_t stream) {
    const int*   sequence     = (const int*)  d_in[0];
    const int*   seq_neighbor = (const int*)  d_in[1];
    const int*   rel_neigh    = (const int*)  d_in[2];
    const float* user_emb     = (const float*)d_in[3];
    const float* rel_emb      = (const float*)d_in[4];
    const float* weight_b     = (const float*)d_in[5];
    float*       out          = (float*)d_out;

    const int blocks = (B_ * L_) / WPB;   // 12800 / 8 = 1600, exact
    subgl_agg_kernel<<<blocks, WPB * 32, 0, stream>>>(
        sequence, seq_neighbor, rel_neigh, user_emb, rel_emb, weight_b, out);
}